// SpatialTransformer_38104949850216
// MI455X (gfx1250) — compile-verified
//
#include <hip/hip_runtime.h>

// 3-D spatial transformer (VoxelMorph dense warp), trilinear interpolation.
// vol:  [B=2, X=160, Y=160, Z=160, C=1]  float32
// flow: [B=2, X=160, Y=160, Z=160, 3]    float32
// out:  [B=2, X=160, Y=160, Z=160, 1]    float32
//
// Memory-bound gather (~165 MB traffic, ~0.25 GFLOP -> ~7 us at 23.3 TB/s).
// Launch geometry carries the coordinates (no integer division):
//   blockIdx = (y, x, b), threadIdx.x = z  -> lanes stream contiguous z.
// Flow: NT b96 loads (read-once stream, keep out of vol's cache set).
// Vol:  RT gather (32 MB working set, ~8x reuse, lives in L2/WGP$) +
//       early identity-position prefetch (flow ~ N(0,1) => sample is within
//       ~1-2 voxels of identity; prefetch overlaps the vol line fetch with
//       the flow load and weight math).
// Out:  NT store (write-once).

#define DIM    160
#define DIMM1  159
#define PLANE  (DIM * DIM)           // 25600
#define VOLSZ  (DIM * DIM * DIM)     // 4096000

__global__ __launch_bounds__(DIM) void spatial_transformer_warp3d(
    const float* __restrict__ vol,
    const float* __restrict__ flow,
    float* __restrict__ out)
{
    // Coordinates directly from launch geometry: zero per-thread divisions.
    const int z = threadIdx.x;   // 0..159 (contiguous along lanes)
    const int y = blockIdx.x;    // 0..159 (block-uniform -> SALU)
    const int x = blockIdx.y;    // 0..159
    const int b = blockIdx.z;    // 0..1

    const float* __restrict__ vb = vol + (size_t)b * VOLSZ;

    // Identity-position index within this batch's volume (uniform + z).
    const int ident = x * PLANE + y * DIM + z;

    // Warm the cache line the gather will almost certainly hit
    // (global_prefetch_b8, WGP scope -> pulls into all cache levels).
    __builtin_prefetch(&vb[ident], 0, 3);

    // ---- streamed flow read (read-once): NT b96 load ----
    const int n = ((b * DIM + x) * DIM + y) * DIM + z;   // output linear index
    const size_t f = (size_t)n * 3;
    const float fx = __builtin_nontemporal_load(&flow[f + 0]);
    const float fy = __builtin_nontemporal_load(&flow[f + 1]);
    const float fz = __builtin_nontemporal_load(&flow[f + 2]);

    // ---- absolute sample location, clipped to [0, dim-1] ----
    const float lx = fminf(fmaxf((float)x + fx, 0.0f), (float)DIMM1);
    const float ly = fminf(fmaxf((float)y + fy, 0.0f), (float)DIMM1);
    const float lz = fminf(fmaxf((float)z + fz, 0.0f), (float)DIMM1);

    // ---- corner indices (reference: floor(clipped), hi = min(lo+1, max)) ----
    const int x0 = (int)floorf(lx);
    const int y0 = (int)floorf(ly);
    const int z0 = (int)floorf(lz);
    const int x1 = min(x0 + 1, DIMM1);
    const int y1 = min(y0 + 1, DIMM1);
    const int z1 = min(z0 + 1, DIMM1);

    // ---- voxelmorph weights: corner-0 weight = float(loc1) - clipped_loc ----
    const float wlx = (float)x1 - lx;  const float whx = 1.0f - wlx;
    const float wly = (float)y1 - ly;  const float why = 1.0f - wly;
    const float wlz = (float)z1 - lz;  const float whz = 1.0f - wlz;

    // ---- 8-corner gather (RT cached: vol is the reusable working set) ----
    const int px0 = x0 * PLANE, px1 = x1 * PLANE;
    const int ry0 = y0 * DIM,   ry1 = y1 * DIM;

    const float v000 = vb[px0 + ry0 + z0];
    const float v001 = vb[px0 + ry0 + z1];
    const float v010 = vb[px0 + ry1 + z0];
    const float v011 = vb[px0 + ry1 + z1];
    const float v100 = vb[px1 + ry0 + z0];
    const float v101 = vb[px1 + ry0 + z1];
    const float v110 = vb[px1 + ry1 + z0];
    const float v111 = vb[px1 + ry1 + z1];

    // ---- trilinear blend (separable: reduce z, then y, then x) ----
    const float c00 = v000 * wlz + v001 * whz;
    const float c01 = v010 * wlz + v011 * whz;
    const float c10 = v100 * wlz + v101 * whz;
    const float c11 = v110 * wlz + v111 * whz;
    const float c0  = c00 * wly + c01 * why;
    const float c1  = c10 * wly + c11 * why;
    const float res = c0 * wlx + c1 * whx;

    // ---- write-once output: non-temporal store ----
    __builtin_nontemporal_store(res, &out[n]);
}

extern "C" void kernel_launch(void* const* d_in, const int* in_sizes, int n_in,
                              void* d_out, int out_size, void* d_ws, size_t ws_size,
                              hipStream_t stream)
{
    const float* vol  = (const float*)d_in[0];   // [2,160,160,160,1]
    const float* flow = (const float*)d_in[1];   // [2,160,160,160,3]
    float*       out  = (float*)d_out;           // [2,160,160,160,1]

    // blockIdx = (y, x, b); threadIdx.x = z (160 = 5 wave32 waves).
    dim3 grid(DIM, DIM, 2);
    dim3 block(DIM, 1, 1);
    spatial_transformer_warp3d<<<grid, block, 0, stream>>>(vol, flow, out);
}